// Sampler_ASGCN_90168543412429
// MI455X (gfx1250) — compile-verified
//
#include <hip/hip_runtime.h>
#include <cstdint>
#include <climits>

typedef float v2f __attribute__((ext_vector_type(2)));
typedef float v8f __attribute__((ext_vector_type(8)));

#define TPB 256

// ------------------------------------------------------------------
// K0: zero the small accumulators (count, sum(q1), means[D], lvec[D])
// ------------------------------------------------------------------
__global__ void k_init(int* __restrict__ count, float* __restrict__ sumq,
                       float* __restrict__ means, float* __restrict__ lvec, int D) {
  int t = blockIdx.x * blockDim.x + threadIdx.x;
  if (t == 0) { *count = 0; *sumq = 0.0f; }
  for (int d = t; d < D; d += blockDim.x * gridDim.x) { means[d] = 0.0f; lvec[d] = 0.0f; }
}

// ------------------------------------------------------------------
// K1: fw2[n] = features[n]·w2 , fw1[n] = features[n]·w1
//     One f32 WMMA GEMV: A = 16-row feature tile (16x4 per step),
//     B = [w2 | w1 | 0 ...] (4x16), K-loop over D in steps of 4.
//     B loads are branchless (pointer select + zero mask) so the hot
//     loop has no EXEC save/restore around the WMMA.
// ------------------------------------------------------------------
__global__ void k_fw_wmma(const float* __restrict__ F, const float* __restrict__ w1,
                          const float* __restrict__ w2, float* __restrict__ fw1,
                          float* __restrict__ fw2, int N, int D) {
  const int lane = threadIdx.x & 31;
  const int wave = (blockIdx.x * blockDim.x + threadIdx.x) >> 5;
  const int tile = wave;                 // 16 rows per tile
  if (tile * 16 >= N) return;            // wave-uniform exit (EXEC all-1 inside)
  const int row0 = tile * 16;
  const int m   = lane & 15;             // A row within tile / C column
  const int kh  = lane >> 4;             // K half-select
  const float* frow = F + (size_t)(row0 + m) * (size_t)D;
  const float* wp   = (m == 1) ? w1 : w2;          // lanes m>=2 load w2, masked to 0
  const float  msk  = (m < 2) ? 1.0f : 0.0f;

  v8f c = {};
  for (int k0 = 0; k0 < D; k0 += 4) {
    const int kb = k0 + 2 * kh;
    // A (16x4 f32): lanes 0-15 hold K={k0,k0+1}, lanes 16-31 hold K={k0+2,k0+3}
    v2f a;
    a.x = frow[kb];
    a.y = frow[kb + 1];
    // B (4x16 f32): column 0 = w2, column 1 = w1, rest zero (branchless).
    v2f b;
    b.x = wp[kb] * msk;
    b.y = wp[kb + 1] * msk;
    c = __builtin_amdgcn_wmma_f32_16x16x4_f32(false, a, false, b,
                                              (short)0, c, false, false);
  }
  // C/D 16x16 f32 layout: VGPR i, lane l: row = i + 8*(l>>4), col = l&15
  if (m < 2) {
    float* dst = (m == 0) ? fw2 : fw1;
    const int rbase = row0 + 8 * kh;
#pragma unroll
    for (int i = 0; i < 8; ++i) dst[rbase + i] = c[i];
  }
}

// ------------------------------------------------------------------
// K1b: hw1[b] = fw1[v[b]]
// ------------------------------------------------------------------
__global__ void k_hw1(const float* __restrict__ fw1, const int* __restrict__ v,
                      float* __restrict__ hw1, int B) {
  int b = blockIdx.x * blockDim.x + threadIdx.x;
  if (b < B) hw1[b] = fw1[v[b]];
}

// ------------------------------------------------------------------
// K2: single fused pass over adj_batch (the big 205 MB read):
//   colsum[n], q1[n] = colsum>0 ? sum_b adj[b,n]*relu(hw1[b]+fw2[n]+1) : 0
//   neighbor compaction + count. Division by num_neis deferred.
//   hw1 table staged into LDS via async global->LDS DMA (ASYNCcnt);
//   the strided adj walk is software-prefetched (global_prefetch_b8).
// ------------------------------------------------------------------
__global__ void k_cols(const float* __restrict__ adj, const float* __restrict__ fw2,
                       const float* __restrict__ hw1, float* __restrict__ q1,
                       int* __restrict__ nlist, int* __restrict__ count,
                       int N, int B) {
  __shared__ float sh[1024];
  const int t = threadIdx.x;
  // async copy of hw1[0..B) into LDS, 16B per lane (tracked by ASYNCcnt)
  const int nchunk = B >> 2;                     // float4 chunks
  if (t < nchunk) {
    const unsigned lds_off = (unsigned)(uintptr_t)(&sh[t * 4]);  // LDS byte offset
    const float* gsrc = hw1 + t * 4;
    asm volatile("global_load_async_to_lds_b128 %0, %1, off"
                 :: "v"(lds_off), "v"(gsrc) : "memory");
  }
  for (int b = (nchunk << 2) + t; b < B; b += blockDim.x) sh[b] = hw1[b];
  asm volatile("s_wait_asynccnt 0" ::: "memory");
  __syncthreads();

  const int n = blockIdx.x * blockDim.x + t;
  if (n >= N) return;
  const float cbias = fw2[n] + 1.0f;
  float cs = 0.0f, ws = 0.0f;
  for (int b = 0; b < B; ++b) {
    if (b + 16 < B) __builtin_prefetch(adj + (size_t)(b + 16) * (size_t)N + n, 0, 1);
    const float a = adj[(size_t)b * (size_t)N + n];  // coalesced across threads
    cs += a;
    const float tt = sh[b] + cbias;
    ws += a * (tt > 0.0f ? tt : 0.0f);
  }
  const float q = (cs > 0.0f) ? ws : 0.0f;
  q1[n] = q;
  if (cs > 0.0f) {
    const int idx = atomicAdd(count, 1);
    nlist[idx] = n;
  }
}

// ------------------------------------------------------------------
// K2b: per-neighbor scalars: lp[i] = log(q1/num_neis), sumq = sum q1
// ------------------------------------------------------------------
__global__ void k_neib(const float* __restrict__ q1, const int* __restrict__ countp,
                       const int* __restrict__ nlist, float* __restrict__ lp,
                       float* __restrict__ sumq) {
  __shared__ float red[TPB];
  const int M = *countp;
  const float numf = (float)M;
  float local = 0.0f;
  for (int i = threadIdx.x; i < M; i += TPB) {
    const float q = q1[nlist[i]];
    local += q;
    lp[i] = logf(q / numf);   // log p1 (=-inf when q==0: never sampled)
  }
  red[threadIdx.x] = local;
  __syncthreads();
  for (int off = TPB / 2; off > 0; off >>= 1) {
    if (threadIdx.x < off) red[threadIdx.x] += red[threadIdx.x + off];
    __syncthreads();
  }
  if (threadIdx.x == 0) *sumq = red[0];
}

// ------------------------------------------------------------------
// K3a: means[d] = sum over neighbor rows of features[n,d]
// ------------------------------------------------------------------
__global__ void k_means(const float* __restrict__ F, const int* __restrict__ nlist,
                        const int* __restrict__ countp, float* __restrict__ means,
                        int D) {
  const int M = *countp;
  const int d = threadIdx.x;   // blockDim.x == D
  float acc = 0.0f;
  for (int i = blockIdx.x; i < M; i += gridDim.x)
    acc += F[(size_t)nlist[i] * (size_t)D + d];
  atomicAdd(&means[d], acc);
}

// ------------------------------------------------------------------
// K3b: lvec[d] = sum_n p_u[n] * (f[n,d]-means[d])^2 ; p_u = q1/sum(q1)
// ------------------------------------------------------------------
__global__ void k_lossvec(const float* __restrict__ F, const float* __restrict__ q1,
                          const int* __restrict__ nlist, const int* __restrict__ countp,
                          const float* __restrict__ means, const float* __restrict__ sumq,
                          float* __restrict__ lvec, int D) {
  const int M = *countp;
  const float inv = 1.0f / *sumq;
  const int d = threadIdx.x;   // blockDim.x == D
  const float mn = means[d];
  float acc = 0.0f;
  for (int i = blockIdx.x; i < M; i += gridDim.x) {
    const int n = nlist[i];
    const float pu = q1[n] * inv;
    const float fc = F[(size_t)n * (size_t)D + d] - mn;
    acc += fc * fc * pu;
  }
  atomicAdd(&lvec[d], acc);
}

// ------------------------------------------------------------------
// JAX threefry-2x32, key = threefry_seed(1) = (0, 1)
// ------------------------------------------------------------------
__device__ __forceinline__ uint32_t rotl32(uint32_t x, uint32_t r) {
  return (x << r) | (x >> (32u - r));
}
__device__ __forceinline__ void threefry2x32_key01(uint32_t c0, uint32_t c1,
                                                   uint32_t& o0, uint32_t& o1) {
  const uint32_t ks0 = 0u, ks1 = 1u, ks2 = 0x1BD11BDBu;  // 0^1^0x1BD11BDA
  uint32_t x0 = c0 + ks0;
  uint32_t x1 = c1 + ks1;
#define TF_R(r) { x0 += x1; x1 = rotl32(x1, (r)); x1 ^= x0; }
  TF_R(13) TF_R(15) TF_R(26) TF_R(6)   x0 += ks1; x1 += ks2 + 1u;
  TF_R(17) TF_R(29) TF_R(16) TF_R(24)  x0 += ks2; x1 += ks0 + 2u;
  TF_R(13) TF_R(15) TF_R(26) TF_R(6)   x0 += ks0; x1 += ks1 + 3u;
  TF_R(17) TF_R(29) TF_R(16) TF_R(24)  x0 += ks1; x1 += ks2 + 4u;
  TF_R(13) TF_R(15) TF_R(26) TF_R(6)   x0 += ks2; x1 += ks0 + 5u;
#undef TF_R
  o0 = x0; o1 = x1;
}

// JAX: u = (bitcast(bits>>9 | 0x3f800000) - 1) * (1-tiny) + tiny; max(tiny, u)
//      gumbel = -log(-log(u))     (note (1-tiny) rounds to 1.0f in f32)
__device__ __forceinline__ float bits_to_gumbel(uint32_t bits) {
  const float tiny = 1.1754943508222875e-38f;
  const float f = __uint_as_float((bits >> 9) | 0x3f800000u) - 1.0f;
  const float u = fmaxf(f + tiny, tiny);
  return -logf(-logf(u));
}

// ------------------------------------------------------------------
// K4: categorical sampling. Block s handles samples s and s+S/2:
//     flat gumbel index j = s*N+n pairs with j+half via one threefry
//     block (JAX counter layout). Only neighbor columns evaluated
//     (-inf logits elsewhere can never win). First-index tie-break.
// ------------------------------------------------------------------
__global__ void k_sample(const int* __restrict__ nlist, const float* __restrict__ lp,
                         const int* __restrict__ countp, int N, int S,
                         int* __restrict__ sampled) {
  __shared__ float sv0[TPB], sv1[TPB];
  __shared__ int   si0[TPB], si1[TPB];
  const int M = *countp;
  const int s = blockIdx.x;                          // 0 .. S/2-1
  const uint32_t half = (uint32_t)((uint64_t)(S / 2) * (uint64_t)N);
  float b0 = -__builtin_huge_valf(), b1 = -__builtin_huge_valf();
  int   i0 = INT_MAX, i1 = INT_MAX;
  for (int i = threadIdx.x; i < M; i += TPB) {
    const int n = nlist[i];
    const float l = lp[i];
    const uint32_t c0 = (uint32_t)((uint64_t)s * (uint64_t)N + (uint64_t)n);
    uint32_t r0, r1;
    threefry2x32_key01(c0, c0 + half, r0, r1);
    const float v0 = bits_to_gumbel(r0) + l;  // sample s
    const float v1 = bits_to_gumbel(r1) + l;  // sample s + S/2
    if (v0 > b0 || (v0 == b0 && n < i0)) { b0 = v0; i0 = n; }
    if (v1 > b1 || (v1 == b1 && n < i1)) { b1 = v1; i1 = n; }
  }
  sv0[threadIdx.x] = b0; si0[threadIdx.x] = i0;
  sv1[threadIdx.x] = b1; si1[threadIdx.x] = i1;
  __syncthreads();
  for (int off = TPB / 2; off > 0; off >>= 1) {
    if (threadIdx.x < off) {
      const int t = threadIdx.x, u = t + off;
      if (sv0[u] > sv0[t] || (sv0[u] == sv0[t] && si0[u] < si0[t])) { sv0[t] = sv0[u]; si0[t] = si0[u]; }
      if (sv1[u] > sv1[t] || (sv1[u] == sv1[t] && si1[u] < si1[t])) { sv1[t] = sv1[u]; si1[t] = si1[u]; }
    }
    __syncthreads();
  }
  if (threadIdx.x == 0) {
    sampled[s]         = (si0[0] == INT_MAX) ? 0 : si0[0];
    sampled[s + S / 2] = (si1[0] == INT_MAX) ? 0 : si1[0];
  }
}

// ------------------------------------------------------------------
// K5a: x_u = features[sampled]                    -> out[0 : S*D]
// ------------------------------------------------------------------
__global__ void k_xu(const float* __restrict__ F, const int* __restrict__ sampled,
                     float* __restrict__ out, int S, int D) {
  const int idx = blockIdx.x * blockDim.x + threadIdx.x;
  if (idx >= S * D) return;
  const int s = idx / D, d = idx - s * D;
  out[idx] = F[(size_t)sampled[s] * (size_t)D + d];
}

// ------------------------------------------------------------------
// K5b: support[b,s] = adj[b, sampled[s]] / (p1[sampled[s]] * S)
//      p1 = q1 / num_neis                          -> out[S*D : S*D+B*S]
// ------------------------------------------------------------------
__global__ void k_support(const float* __restrict__ adj, const float* __restrict__ q1,
                          const int* __restrict__ sampled, const int* __restrict__ countp,
                          float* __restrict__ out, int B, int S, int N) {
  const int idx = blockIdx.x * blockDim.x + threadIdx.x;
  if (idx >= B * S) return;
  const int b = idx / S, s = idx - b * S;
  const int n = sampled[s];
  const float p1s = q1[n] / (float)(*countp);
  out[idx] = adj[(size_t)b * (size_t)N + n] / (p1s * (float)S);
}

// ------------------------------------------------------------------
// K5c: loss = mean_d lvec[d]                       -> out[S*D + B*S]
// ------------------------------------------------------------------
__global__ void k_loss(const float* __restrict__ lvec, float* __restrict__ out, int D) {
  __shared__ float red[TPB];
  float acc = 0.0f;
  for (int d = threadIdx.x; d < D; d += TPB) acc += lvec[d];
  red[threadIdx.x] = acc;
  __syncthreads();
  for (int off = TPB / 2; off > 0; off >>= 1) {
    if (threadIdx.x < off) red[threadIdx.x] += red[threadIdx.x + off];
    __syncthreads();
  }
  if (threadIdx.x == 0) *out = red[0] / (float)D;
}

// ------------------------------------------------------------------
extern "C" void kernel_launch(void* const* d_in, const int* in_sizes, int n_in,
                              void* d_out, int out_size, void* d_ws, size_t ws_size,
                              hipStream_t stream) {
  (void)n_in; (void)ws_size;
  const float* F   = (const float*)d_in[0];
  const float* adj = (const float*)d_in[1];
  const float* w1  = (const float*)d_in[2];
  const float* w2  = (const float*)d_in[3];
  const int*   v   = (const int*)d_in[4];

  const int D = in_sizes[2];                 // w1: [D,1]
  const int N = in_sizes[0] / D;             // features: [N,D]
  const int B = in_sizes[4];                 // v: [B]
  const int S = (out_size - 1) / (D + B);    // S*D + B*S + 1 = out_size

  // workspace carve-out (256B aligned chunks)
  char* p = (char*)d_ws;
  auto carve = [&](size_t bytes) -> void* {
    void* r = (void*)p;
    p += (bytes + 255) & ~(size_t)255;
    return r;
  };
  float* fw1   = (float*)carve((size_t)N * 4);
  float* fw2   = (float*)carve((size_t)N * 4);
  float* q1    = (float*)carve((size_t)N * 4);
  int*   nlist = (int*)  carve((size_t)N * 4);
  float* lp    = (float*)carve((size_t)N * 4);
  float* hw1   = (float*)carve((size_t)B * 4);
  float* means = (float*)carve((size_t)D * 4);
  float* lvec  = (float*)carve((size_t)D * 4);
  int*   smp   = (int*)  carve((size_t)S * 4);
  int*   count = (int*)  carve(64);
  float* sumq  = (float*)carve(64);

  float* out_xu      = (float*)d_out;
  float* out_support = out_xu + (size_t)S * D;
  float* out_loss    = out_support + (size_t)B * S;

  // 0) zero accumulators
  k_init<<<1, TPB, 0, stream>>>(count, sumq, means, lvec, D);

  // 1) WMMA GEMV: fw2/fw1 (one pass over 205 MB of features)
  const int tiles = (N + 15) / 16;                  // 16 rows per wave
  const int fw_blocks = (tiles + 7) / 8;            // 8 waves per 256-thread block
  k_fw_wmma<<<fw_blocks, TPB, 0, stream>>>(F, w1, w2, fw1, fw2, N, D);

  // 1b) hw1 gather
  k_hw1<<<(B + TPB - 1) / TPB, TPB, 0, stream>>>(fw1, v, hw1, B);

  // 2) fused adj pass: colsum + q1 + neighbor compaction (one 205 MB pass)
  k_cols<<<(N + TPB - 1) / TPB, TPB, 0, stream>>>(adj, fw2, hw1, q1, nlist, count, N, B);

  // 2b) neighbor scalars
  k_neib<<<1, TPB, 0, stream>>>(q1, count, nlist, lp, sumq);

  // 3) loss statistics (two cheap passes over neighbor rows)
  k_means<<<120, D, 0, stream>>>(F, nlist, count, means, D);
  k_lossvec<<<120, D, 0, stream>>>(F, q1, nlist, count, means, sumq, lvec, D);

  // 4) gumbel-argmax categorical sampling (JAX threefry, key(1))
  k_sample<<<S / 2, TPB, 0, stream>>>(nlist, lp, count, N, S, smp);

  // 5) outputs
  k_xu<<<((S * D) + TPB - 1) / TPB, TPB, 0, stream>>>(F, smp, out_xu, S, D);
  k_support<<<((B * S) + TPB - 1) / TPB, TPB, 0, stream>>>(adj, q1, smp, count,
                                                           out_support, B, S, N);
  k_loss<<<1, TPB, 0, stream>>>(lvec, out_loss, D);
}